// GCNBlock_72052371357885
// MI455X (gfx1250) — compile-verified
//
#include <hip/hip_runtime.h>

// ---------------------------------------------------------------------------
// GCN block: edge-dropout GCN conv (norm='both') + BatchNorm + node dropout
//   h   = features @ W     (WMMA f32 16x16x4; W staged in LDS via async DMA)
//   agg = segsum(h[src]*coef, dst)           (L2-resident float atomics)
//   y   = BN(agg)*gamma+beta, * node mask    (bias b cancels inside BN)
// ---------------------------------------------------------------------------

#define N_NODES  50000
#define N_EDGES  800000
#define IN_FEAT  256
#define OUT_FEAT 128
#define P_EDGE   0.2f
#define P_NODE   0.1f
#define BN_EPS   1e-5f

// LDS row stride for W: 128 + 8 pad floats. 2*136 = 272 == 16 (mod 64), so the
// two half-waves (K and K+2 rows) hit disjoint 16-bank sets -> conflict-free.
#define W_LDS_STRIDE 136

typedef __attribute__((ext_vector_type(2))) float v2f;
typedef __attribute__((ext_vector_type(8))) float v8f;

// ---------------- zero workspace ----------------
__global__ void __launch_bounds__(256) zero_kernel(float* __restrict__ p, long n) {
    long i = (long)blockIdx.x * blockDim.x + threadIdx.x;
    long stride = (long)gridDim.x * blockDim.x;
    for (; i < n; i += stride) p[i] = 0.0f;
}

// ---------------- edge degrees (dropped graph) ----------------
__global__ void __launch_bounds__(256) degree_kernel(const int* __restrict__ src,
                                                     const int* __restrict__ dst,
                                                     const float* __restrict__ erand,
                                                     float* __restrict__ deg_src,
                                                     float* __restrict__ deg_dst) {
    int e = blockIdx.x * blockDim.x + threadIdx.x;
    if (e >= N_EDGES) return;
    if (erand[e] >= P_EDGE) {
        atomicAdd(&deg_src[src[e]], 1.0f);
        atomicAdd(&deg_dst[dst[e]], 1.0f);
    }
}

// ---------------- per-edge normalization coefficient ----------------
__global__ void __launch_bounds__(256) coef_kernel(const int* __restrict__ src,
                                                   const int* __restrict__ dst,
                                                   const float* __restrict__ erand,
                                                   const float* __restrict__ deg_src,
                                                   const float* __restrict__ deg_dst,
                                                   float* __restrict__ coef) {
    int e = blockIdx.x * blockDim.x + threadIdx.x;
    if (e >= N_EDGES) return;
    float m  = (erand[e] >= P_EDGE) ? 1.0f : 0.0f;
    float ds = fmaxf(deg_src[src[e]], 1.0f);
    float dd = fmaxf(deg_dst[dst[e]], 1.0f);
    coef[e] = m * rsqrtf(ds * dd);
}

// ---------------- h = features @ W via V_WMMA_F32_16X16X4_F32 ----------------
// Block = 8 waves; wave w owns output tile (mtile, ntile=w). 50000 = 3125*16,
// 128 = 8*16 -> exact tiling, EXEC stays all-ones (WMMA requirement).
// W (256x128 f32, 128 KB) is staged once per block into bank-padded LDS with
// global_load_async_to_lds_b128 (ASYNCcnt), then B fragments come from DS.
__global__ void __launch_bounds__(256) gemm_wmma_kernel(const float* __restrict__ A,  // [N,256]
                                                        const float* __restrict__ W,  // [256,128]
                                                        float* __restrict__ H) {      // [N,128]
    __shared__ float wlds[IN_FEAT * W_LDS_STRIDE];   // 136 KB, 2 blocks/WGP

    // ---- async stage W into LDS: 8192 16B chunks, 32 per thread ----
    {
        unsigned lbase = (unsigned)(size_t)(__attribute__((address_space(3))) float*)wlds;
        #pragma unroll 4
        for (int i = 0; i < 32; ++i) {
            const int c  = (int)threadIdx.x + 256 * i;   // 16B-chunk id
            const int r  = c >> 5;                        // W row   (0..255)
            const int c4 = c & 31;                        // chunk within row
            unsigned loff = lbase + (unsigned)(r * W_LDS_STRIDE + c4 * 4) * 4u;
            const float* g = W + (size_t)c * 4;
            asm volatile("global_load_async_to_lds_b128 %0, %1, off"
                         :: "v"(loff), "v"(g) : "memory");
        }
        asm volatile("s_wait_asynccnt 0x0" ::: "memory");
        __syncthreads();
    }

    const int lane  = threadIdx.x & 31;
    const int wave  = threadIdx.x >> 5;          // ntile 0..7
    const int mtile = blockIdx.x;                // 0..3124
    const int mloc  = lane & 15;
    const int n     = wave * 16 + mloc;          // output column
    const int ksel  = (lane >> 4) << 1;          // 0 (lanes 0-15) or 2 (lanes 16-31)

    const float* __restrict__ arow = A + (long)(mtile * 16 + mloc) * IN_FEAT + ksel;
    const float* __restrict__ wrow = wlds + (long)ksel * W_LDS_STRIDE + n;

    v8f c = {};
    #pragma unroll 4
    for (int k0 = 0; k0 < IN_FEAT; k0 += 4) {
        // A fragment 16x4: lane holds {K=k0+ksel, K=k0+ksel+1} of row mloc
        v2f a = *(const v2f*)(arow + k0);
        // B fragment 4x16 from LDS: lane holds {K=k0+ksel, K=k0+ksel+1}, col n
        v2f b;
        b.x = wrow[k0 * W_LDS_STRIDE];
        b.y = wrow[k0 * W_LDS_STRIDE + W_LDS_STRIDE];
        c = __builtin_amdgcn_wmma_f32_16x16x4_f32(
                /*neg_a=*/false, a, /*neg_b=*/false, b,
                /*c_mod=*/(short)0, c, /*reuse_a=*/false, /*reuse_b=*/false);
    }

    // C/D layout: VGPR i -> row (i + 8*(lane>=16)), col = lane&15
    const int row0 = mtile * 16 + ((lane >> 4) << 3);
    float* __restrict__ hout = H + (long)row0 * OUT_FEAT + n;
    #pragma unroll
    for (int i = 0; i < 8; ++i) hout[(long)i * OUT_FEAT] = c[i];
}

// ---------------- scatter: agg[dst] += h[src] * coef ----------------
// One wave per edge (uniform EXEC/branches), lane covers 4 features:
// one b128 gather + 4 global f32 atomics; whole working set is L2-resident.
__global__ void __launch_bounds__(256) scatter_kernel(const int* __restrict__ src,
                                                      const int* __restrict__ dst,
                                                      const float* __restrict__ coef,
                                                      const float* __restrict__ H,
                                                      float* __restrict__ agg) {
    const int lane = threadIdx.x & 31;
    const int e    = blockIdx.x * 8 + (threadIdx.x >> 5);
    if (e >= N_EDGES) return;
    const float cf = coef[e];
    if (cf == 0.0f) return;                       // dropped edge: wave-uniform skip
    const int s = src[e], d = dst[e];
    const float4 hv = *(const float4*)(H + (long)s * OUT_FEAT + lane * 4);
    float* __restrict__ aout = agg + (long)d * OUT_FEAT + lane * 4;
    atomicAdd(aout + 0, hv.x * cf);
    atomicAdd(aout + 1, hv.y * cf);
    atomicAdd(aout + 2, hv.z * cf);
    atomicAdd(aout + 3, hv.w * cf);
}

// ---------------- BN column statistics (sum, sumsq) ----------------
#define ROWS_PER_BLOCK 250
__global__ void __launch_bounds__(128) colstats_kernel(const float* __restrict__ agg,
                                                       float* __restrict__ sum,
                                                       float* __restrict__ sumsq) {
    const int f    = threadIdx.x;                 // feature column 0..127
    const int row0 = blockIdx.x * ROWS_PER_BLOCK;
    float s = 0.0f, s2 = 0.0f;
    #pragma unroll 2
    for (int r = row0; r < row0 + ROWS_PER_BLOCK; ++r) {
        float v = agg[(long)r * OUT_FEAT + f];
        s += v; s2 += v * v;
    }
    atomicAdd(&sum[f], s);
    atomicAdd(&sumsq[f], s2);
}

// ---------------- BN normalize + node-feature dropout ----------------
__global__ void __launch_bounds__(256) finalize_kernel(const float* __restrict__ agg,
                                                       const float* __restrict__ sum,
                                                       const float* __restrict__ sumsq,
                                                       const float* __restrict__ gamma,
                                                       const float* __restrict__ beta,
                                                       const float* __restrict__ nrand,
                                                       float* __restrict__ out) {
    long i = (long)blockIdx.x * blockDim.x + threadIdx.x;
    if (i >= (long)N_NODES * OUT_FEAT) return;
    const int f = (int)(i & (OUT_FEAT - 1));
    const float invN = 1.0f / (float)N_NODES;
    const float mean = sum[f] * invN;
    const float var  = fmaxf(sumsq[f] * invN - mean * mean, 0.0f);
    const float rstd = rsqrtf(var + BN_EPS);
    const float keep = (nrand[i] >= P_NODE) ? (1.0f / (1.0f - P_NODE)) : 0.0f;
    out[i] = ((agg[i] - mean) * rstd * gamma[f] + beta[f]) * keep;
}

// ---------------------------------------------------------------------------
extern "C" void kernel_launch(void* const* d_in, const int* in_sizes, int n_in,
                              void* d_out, int out_size, void* d_ws, size_t ws_size,
                              hipStream_t stream) {
    const float* features = (const float*)d_in[0];
    const float* W        = (const float*)d_in[1];
    // d_in[2] = b: a constant per-column shift cancels exactly inside BatchNorm
    // ((x+b) - mean(x+b) == x - mean(x); var unchanged), so it is unused.
    const float* gamma    = (const float*)d_in[3];
    const float* beta     = (const float*)d_in[4];
    const int*   src      = (const int*)d_in[5];
    const int*   dst      = (const int*)d_in[6];
    const float* erand    = (const float*)d_in[7];
    const float* nrand    = (const float*)d_in[8];
    float*       out      = (float*)d_out;

    // workspace layout (floats)
    float* ws      = (float*)d_ws;
    float* h       = ws;                                    // N*128
    float* agg     = h       + (size_t)N_NODES * OUT_FEAT;  // N*128
    float* deg_src = agg     + (size_t)N_NODES * OUT_FEAT;  // N
    float* deg_dst = deg_src + N_NODES;                     // N
    float* sum     = deg_dst + N_NODES;                     // 128
    float* sumsq   = sum     + OUT_FEAT;                    // 128
    float* coef    = sumsq   + OUT_FEAT;                    // E

    // zero agg..sumsq (contiguous)
    const long zcount = (long)N_NODES * OUT_FEAT + 2L * N_NODES + 2L * OUT_FEAT;
    zero_kernel<<<2048, 256, 0, stream>>>(agg, zcount);

    degree_kernel<<<(N_EDGES + 255) / 256, 256, 0, stream>>>(src, dst, erand, deg_src, deg_dst);
    coef_kernel<<<(N_EDGES + 255) / 256, 256, 0, stream>>>(src, dst, erand, deg_src, deg_dst, coef);

    gemm_wmma_kernel<<<N_NODES / 16, 256, 0, stream>>>(features, W, h);

    scatter_kernel<<<N_EDGES / 8, 256, 0, stream>>>(src, dst, coef, h, agg);

    colstats_kernel<<<N_NODES / ROWS_PER_BLOCK, 128, 0, stream>>>(agg, sum, sumsq);

    const long total = (long)N_NODES * OUT_FEAT;
    finalize_kernel<<<(unsigned)((total + 255) / 256), 256, 0, stream>>>(
        agg, sum, sumsq, gamma, beta, nrand, out);
}